// Voxelization_63402307224074
// MI455X (gfx1250) — compile-verified
//
#include <hip/hip_runtime.h>
#include <hip/hip_bf16.h>

// ---------------- problem constants (match reference exactly) ----------------
constexpr int   NXC = 442, NYC = 502, NZC = 3;
constexpr int   CELLS = NXC * NYC * NZC;          // 665652
constexpr int   MAXV = 40000;                     // MAX_VOXELS
constexpr int   MAXP = 32;                        // MAX_POINTS
constexpr int   SCAN_BLK = 1024;                  // cells per scan block
constexpr int   NBLK = (CELLS + SCAN_BLK - 1) / SCAN_BLK;   // 651
constexpr int   SCAN_CELLS = NBLK * SCAN_BLK;     // 666624 (padded, counts zeroed)
constexpr unsigned INVALID_V = 0xFFFFFFFFu;

constexpr int PILLAR_ELEMS = MAXV * MAXP * 4;     // 5,120,000 floats
constexpr int COORD_ELEMS  = MAXV * 3;            // 120,000
constexpr int NUM_ELEMS    = MAXV;                // 40,000
constexpr int OUT_TOTAL    = PILLAR_ELEMS + COORD_ELEMS + NUM_ELEMS; // 5,280,000

static_assert(CELLS == 665652, "grid dims");

// ---------------- CDNA5 async-to-LDS plumbing ----------------
#if defined(__has_builtin)
#  if __has_builtin(__builtin_amdgcn_global_load_async_to_lds_b128)
#    define HAVE_ASYNC_BUILTIN 1
#  endif
#  if __has_builtin(__builtin_amdgcn_s_wait_asynccnt)
#    define HAVE_ASYNCCNT_BUILTIN 1
#  endif
#endif

typedef int v4i __attribute__((ext_vector_type(4)));
typedef __attribute__((address_space(1))) v4i* gptr_v4i;   // global (clang prints as __device__)
typedef __attribute__((address_space(3))) v4i* lptr_v4i;   // LDS    (clang prints as __shared__)

// async copy of 16 bytes: global -> LDS (tracked by ASYNCcnt, not LOADcnt)
__device__ __forceinline__ void async_copy16(const void* g, void* l) {
#if defined(HAVE_ASYNC_BUILTIN)
    __builtin_amdgcn_global_load_async_to_lds_b128((gptr_v4i)g, (lptr_v4i)l, 0, 0);
#else
    unsigned lds_off = (unsigned)(size_t)(__attribute__((address_space(3))) void*)l;
    asm volatile("global_load_async_to_lds_b128 %0, %1, off"
                 :: "v"(lds_off), "v"(g) : "memory");
#endif
}

__device__ __forceinline__ void wait_async0() {
#if defined(HAVE_ASYNCCNT_BUILTIN)
    __builtin_amdgcn_s_wait_asynccnt(0);
#else
    asm volatile("s_wait_asynccnt 0" ::: "memory");
#endif
}

// Per-point cell id, bit-exact vs reference: f32 division, floorf, int32 cast,
// inclusive range test on both ends.
__device__ __forceinline__ bool point_cell(const float4 p, int& lin) {
    const float x = p.x, y = p.y, z = p.z;
    bool in = (x >= 0.0f)  & (x <= 70.4f) &
              (y >= -40.0f) & (y <= 40.0f) &
              (z >= -3.0f)  & (z <= 1.0f);
    if (!in) return false;
    int cx = (int)floorf(x / 0.16f);
    int cy = (int)floorf(y / 0.16f);
    int cz = (int)floorf(z / 4.0f);
    int rx = cx - 0;          // _CMIN = (0, -250, -1)
    int ry = cy + 250;
    int rz = cz + 1;
    lin = (rx * NYC + ry) * NZC + rz;
    return true;
}

// ---------------- kernels ----------------

// Histogram pass: stream points through LDS via CDNA5 async-to-LDS DMA path.
// Each lane copies its own 16B point global->LDS (one ASYNC op per wave per chunk),
// waits ASYNCcnt==0, reads back via ds_load_b128, then does an L2-resident atomic.
// Lanes only ever consume their own LDS slot, so no barriers are required; the
// atomic's data-dependency on the ds_load orders the LDS read before the next
// iteration's async overwrite.
__global__ void count_kernel(const float4* __restrict__ pc, int n,
                             unsigned* __restrict__ cellCount) {
    __shared__ float4 buf[256];
    const int tid = threadIdx.x;
    const int nChunks = (n + 255) / 256;
    for (int ch = blockIdx.x; ch < nChunks; ch += gridDim.x) {
        const int i = ch * 256 + tid;
        if (i < n) {
            async_copy16(&pc[i], &buf[tid]);   // global_load_async_to_lds_b128
        }
        wait_async0();                          // s_wait_asynccnt 0
        if (i < n) {
            float4 p = buf[tid];                // ds_load_b128
            int lin;
            if (point_cell(p, lin)) atomicAdd(&cellCount[lin], 1u);
        }
    }
}

__global__ void init_out_kernel(float* __restrict__ out) {
    int i = blockIdx.x * blockDim.x + threadIdx.x;
    if (i >= OUT_TOTAL) return;
    float v = 0.0f;
    if (i >= PILLAR_ELEMS && i < PILLAR_ELEMS + COORD_ELEMS) v = -1.0f; // coords default
    out[i] = v;
}

__global__ void init_ws_kernel(unsigned* __restrict__ cellCount,
                               unsigned* __restrict__ cursor,
                               unsigned* __restrict__ uniq) {
    int i = blockIdx.x * blockDim.x + threadIdx.x;
    if (i < SCAN_CELLS) { cellCount[i] = 0u; cursor[i] = 0u; }
    if (i < MAXV) uniq[i] = INVALID_V;
}

// Pass 1: per-1024-cell block totals of (occupancy, count)
__global__ void scan_pass1(const unsigned* __restrict__ cellCount,
                           unsigned* __restrict__ blkOcc,
                           unsigned* __restrict__ blkCnt) {
    __shared__ unsigned sOcc[256];
    __shared__ unsigned sCnt[256];
    const int b = blockIdx.x, t = threadIdx.x;
    const int base = b * SCAN_BLK + t * 4;
    uint4 c4 = *(const uint4*)&cellCount[base];
    unsigned occ = (c4.x > 0u) + (c4.y > 0u) + (c4.z > 0u) + (c4.w > 0u);
    unsigned cnt = c4.x + c4.y + c4.z + c4.w;
    sOcc[t] = occ; sCnt[t] = cnt;
    __syncthreads();
    for (int s = 128; s > 0; s >>= 1) {
        if (t < s) { sOcc[t] += sOcc[t + s]; sCnt[t] += sCnt[t + s]; }
        __syncthreads();
    }
    if (t == 0) { blkOcc[b] = sOcc[0]; blkCnt[b] = sCnt[0]; }
}

// Pass 2: serial exclusive scan over 651 block totals (trivial work, deterministic)
__global__ void scan_pass2(unsigned* __restrict__ blkOcc, unsigned* __restrict__ blkCnt) {
    if (blockIdx.x == 0 && threadIdx.x == 0) {
        unsigned ro = 0, rc = 0;
        for (int b = 0; b < NBLK; ++b) {
            unsigned o = blkOcc[b], c = blkCnt[b];
            blkOcc[b] = ro; blkCnt[b] = rc;
            ro += o; rc += c;
        }
    }
}

// Pass 3: full exclusive prefixes; emit uniq / cell2voxel / pointStart and the
// pillar_coords + num_points outputs for the first 40000 occupied cells (ascending
// cell id == jnp.unique's sorted+truncated behavior).
__global__ void scan_pass3(const unsigned* __restrict__ cellCount,
                           const unsigned* __restrict__ blkOcc,
                           const unsigned* __restrict__ blkCnt,
                           unsigned* __restrict__ cell2voxel,
                           unsigned* __restrict__ pointStart,
                           unsigned* __restrict__ uniq,
                           float* __restrict__ outCoords,
                           float* __restrict__ outNum) {
    __shared__ unsigned sOcc[256];
    __shared__ unsigned sCnt[256];
    const int b = blockIdx.x, t = threadIdx.x;
    const int base = b * SCAN_BLK + t * 4;
    uint4 c4 = *(const uint4*)&cellCount[base];
    unsigned occs[4] = { c4.x > 0u, c4.y > 0u, c4.z > 0u, c4.w > 0u };
    unsigned cnts[4] = { c4.x, c4.y, c4.z, c4.w };
    unsigned occSum = occs[0] + occs[1] + occs[2] + occs[3];
    unsigned cntSum = cnts[0] + cnts[1] + cnts[2] + cnts[3];
    sOcc[t] = occSum; sCnt[t] = cntSum;
    __syncthreads();
    // inclusive Hillis-Steele scan over 256 thread groups (ds_* traffic)
    for (int s = 1; s < 256; s <<= 1) {
        unsigned vo = 0, vc = 0;
        if (t >= s) { vo = sOcc[t - s]; vc = sCnt[t - s]; }
        __syncthreads();
        if (t >= s) { sOcc[t] += vo; sCnt[t] += vc; }
        __syncthreads();
    }
    unsigned occEx = blkOcc[b] + sOcc[t] - occSum;  // global exclusive prefix at this group
    unsigned cntEx = blkCnt[b] + sCnt[t] - cntSum;
    #pragma unroll
    for (int j = 0; j < 4; ++j) {
        const int c = base + j;
        if (c < CELLS) {
            pointStart[c] = cntEx;
            unsigned mapped = INVALID_V;
            if (cnts[j] > 0u && occEx < (unsigned)MAXV) {
                const unsigned v = occEx;
                mapped = v;
                uniq[v] = (unsigned)c;
                const int cz = c % NZC;
                const int cy = (c / NZC) % NYC;
                const int cx = c / (NZC * NYC);
                outCoords[v * 3 + 0] = (float)(cx + 0);
                outCoords[v * 3 + 1] = (float)(cy - 250);
                outCoords[v * 3 + 2] = (float)(cz - 1);
                // reference quirk: num_points = maximum(count, 32)
                outNum[v] = (float)(cnts[j] < (unsigned)MAXP ? (unsigned)MAXP : cnts[j]);
            }
            cell2voxel[c] = mapped;
        }
        occEx += occs[j];
        cntEx += cnts[j];
    }
}

// Counting-sort scatter of point indices into per-cell lists (kept cells only).
// Direct global_load_b128 + global_prefetch_b8 streaming path.
__global__ void scatter_kernel(const float4* __restrict__ pc, int n,
                               const unsigned* __restrict__ cell2voxel,
                               const unsigned* __restrict__ pointStart,
                               unsigned* __restrict__ cursor,
                               unsigned* __restrict__ sortedIdx) {
    int i = blockIdx.x * blockDim.x + threadIdx.x;
    const int stride = gridDim.x * blockDim.x;
    for (; i < n; i += stride) {
        __builtin_prefetch(&pc[i + stride], 0, 1);   // global_prefetch_b8 (speculative)
        float4 p = pc[i];
        int lin;
        if (point_cell(p, lin)) {
            if (cell2voxel[lin] != INVALID_V) {
                unsigned slot = pointStart[lin] + atomicAdd(&cursor[lin], 1u);
                sortedIdx[slot] = (unsigned)i;
            }
        }
    }
}

// One thread per voxel: pick the <=32 smallest point indices (ascending) from its
// list — reproduces the reference's stable original-index ordering deterministically.
__global__ void fill_kernel(const float4* __restrict__ pc,
                            const unsigned* __restrict__ uniq,
                            const unsigned* __restrict__ cellCount,
                            const unsigned* __restrict__ pointStart,
                            const unsigned* __restrict__ sortedIdx,
                            float4* __restrict__ pillars) {
    const int v = blockIdx.x * blockDim.x + threadIdx.x;
    if (v >= MAXV) return;
    const unsigned c = uniq[v];
    if (c == INVALID_V) return;
    const unsigned count = cellCount[c];
    const unsigned base  = pointStart[c];
    const unsigned m = count < (unsigned)MAXP ? count : (unsigned)MAXP;
    long prev = -1;
    for (unsigned r = 0; r < m; ++r) {
        unsigned best = INVALID_V;
        for (unsigned j = 0; j < count; ++j) {
            unsigned idx = sortedIdx[base + j];
            if ((long)idx > prev && idx < best) best = idx;
        }
        prev = (long)best;
        pillars[(size_t)v * MAXP + r] = pc[best];   // global_load_b128 + global_store_b128
    }
}

// ---------------- launch ----------------
extern "C" void kernel_launch(void* const* d_in, const int* in_sizes, int n_in,
                              void* d_out, int out_size, void* d_ws, size_t ws_size,
                              hipStream_t stream) {
    const float4* pc = (const float4*)d_in[0];
    const int n = in_sizes[0] / 4;                 // 2,000,000 points x 4 channels
    float* out = (float*)d_out;

    // carve workspace (~19 MB), 256B-aligned regions
    char* w = (char*)d_ws;
    auto carve = [&](size_t bytes) -> void* {
        void* p = (void*)w;
        w += (bytes + 255) & ~(size_t)255;
        return p;
    };
    unsigned* cellCount  = (unsigned*)carve((size_t)SCAN_CELLS * 4);
    unsigned* cursor     = (unsigned*)carve((size_t)SCAN_CELLS * 4);
    unsigned* cell2voxel = (unsigned*)carve((size_t)CELLS * 4);
    unsigned* pointStart = (unsigned*)carve((size_t)CELLS * 4);
    unsigned* uniq       = (unsigned*)carve((size_t)MAXV * 4);
    unsigned* blkOcc     = (unsigned*)carve((size_t)NBLK * 4);
    unsigned* blkCnt     = (unsigned*)carve((size_t)NBLK * 4);
    unsigned* sortedIdx  = (unsigned*)carve((size_t)n * 4);

    init_out_kernel<<<(OUT_TOTAL + 255) / 256, 256, 0, stream>>>(out);
    init_ws_kernel<<<(SCAN_CELLS + 255) / 256, 256, 0, stream>>>(cellCount, cursor, uniq);
    count_kernel<<<2048, 256, 0, stream>>>(pc, n, cellCount);
    scan_pass1<<<NBLK, 256, 0, stream>>>(cellCount, blkOcc, blkCnt);
    scan_pass2<<<1, 32, 0, stream>>>(blkOcc, blkCnt);
    scan_pass3<<<NBLK, 256, 0, stream>>>(cellCount, blkOcc, blkCnt, cell2voxel, pointStart,
                                         uniq, out + PILLAR_ELEMS,
                                         out + PILLAR_ELEMS + COORD_ELEMS);
    scatter_kernel<<<4096, 256, 0, stream>>>(pc, n, cell2voxel, pointStart, cursor, sortedIdx);
    fill_kernel<<<(MAXV + 255) / 256, 256, 0, stream>>>(pc, uniq, cellCount, pointStart,
                                                        sortedIdx, (float4*)out);
}